// TemporalGNN_30674656428555
// MI455X (gfx1250) — compile-verified
//
#include <hip/hip_runtime.h>
#include <math.h>

// ---------------------------------------------------------------------------
// TemporalGNN (3x GAT + BN/ELU) for MI455X (gfx1250, wave32).
// Dense node transforms use V_WMMA_F32_16X16X4_F32 (full f32 precision; the
// workload is L2/atomic bound, so there is no roofline reason to drop to f16).
// Graph aggregation uses the identity  softmax-weighted-sum = (sum e_i x_j)/(sum e_i)
// so segment softmax becomes: atomicMax (ordered-uint encoded) -> exp+atomicAdd
// -> per-node normalize fused with bias+BN+ELU.
// ---------------------------------------------------------------------------

#define EDGEDIM 32

typedef __attribute__((ext_vector_type(2))) float v2f;
typedef __attribute__((ext_vector_type(4))) float v4f;
typedef __attribute__((ext_vector_type(8))) float v8f;

// Monotonic float <-> uint mapping so atomicMax on u32 orders floats correctly.
__device__ __forceinline__ unsigned enc_f32(float f) {
  unsigned u = __float_as_uint(f);
  return (u & 0x80000000u) ? ~u : (u | 0x80000000u);
}
__device__ __forceinline__ float dec_f32(unsigned u) {
  u = (u & 0x80000000u) ? (u & 0x7fffffffu) : ~u;
  return __uint_as_float(u);
}

__global__ void k_fill_f32(float* __restrict__ p, float v, int n) {
  int i = blockIdx.x * blockDim.x + threadIdx.x;
  if (i < n) p[i] = v;
}
__global__ void k_fill_u32(unsigned* __restrict__ p, unsigned v, int n) {
  int i = blockIdx.x * blockDim.x + threadIdx.x;
  if (i < n) p[i] = v;
}

// in-degree count + segment-sum of edge_attr over dst (for self-loop attrs)
__global__ void k_degree(const int* __restrict__ dst, const float* __restrict__ edge_attr,
                         float* __restrict__ cnt, float* __restrict__ ea_sum, int E) {
  int e = blockIdx.x * blockDim.x + threadIdx.x;
  if (e >= E) return;
  int d = dst[e];
  atomicAdd(&cnt[d], 1.0f);
  const float* ea = edge_attr + (long long)e * EDGEDIM;
  float* out = ea_sum + (long long)d * EDGEDIM;
#pragma unroll
  for (int k = 0; k < EDGEDIM; ++k) atomicAdd(&out[k], ea[k]);
}

// loop_attr = cnt>0 ? ea_sum/max(cnt,1) : 0   (in place over ea_sum)
__global__ void k_loop_attr(float* __restrict__ loop_attr, const float* __restrict__ cnt, int N) {
  int i = blockIdx.x * blockDim.x + threadIdx.x;
  if (i >= N * EDGEDIM) return;
  float c = cnt[i / EDGEDIM];
  loop_attr[i] = (c > 0.f) ? loop_attr[i] / fmaxf(c, 1.f) : 0.f;
}

// H[N,M] = X[N,K] @ W[K,M], f32, via V_WMMA_F32_16X16X4_F32.
// One wave per 16x16 output tile; tile index is wave-uniform -> EXEC all-1s.
// Requires N%16==0 (50000 ok), M%16==0, K%4==0.
__global__ void k_gemm_wmma(const float* __restrict__ X, const float* __restrict__ W,
                            float* __restrict__ H, int Nrows, int K, int M) {
  const int lane  = threadIdx.x & 31;
  const int wave  = threadIdx.x >> 5;
  const int tilesM = M >> 4;
  const int tile  = blockIdx.x * (blockDim.x >> 5) + wave;
  const int numTiles = (Nrows >> 4) * tilesM;
  if (tile >= numTiles) return;          // wave-uniform branch
  const int tm = tile % tilesM;
  const int tn = tile / tilesM;
  // A 16x4 layout: lanes 0-15 rows 0-15 hold K={k,k+1}; lanes 16-31 hold K={k+2,k+3}
  const int row   = (tn << 4) + (lane & 15);
  const int colB  = (tm << 4) + (lane & 15);
  const int khalf = (lane >> 4) << 1;    // 0 or 2
  const float* xp = X + (long long)row * K + khalf;
  const float* wp = W + (long long)khalf * M + colB;
  v8f c = {0.f, 0.f, 0.f, 0.f, 0.f, 0.f, 0.f, 0.f};
  for (int k = 0; k < K; k += 4) {
    v2f a = *(const v2f*)(xp + k);                                   // 8B aligned
    v2f b = {wp[(long long)k * M], wp[(long long)(k + 1) * M]};
    c = __builtin_amdgcn_wmma_f32_16x16x4_f32(false, a, false, b, (short)0, c, false, false);
  }
  // C/D layout: VGPR j -> row (j + 8*(lane>=16)), col = lane&15
  const int rbase = (tn << 4) + ((lane >> 4) << 3);
  const int col   = (tm << 4) + (lane & 15);
#pragma unroll
  for (int j = 0; j < 8; ++j)
    H[(long long)(rbase + j) * M + col] = c[j];
}

// per (node, head): ai = h . att_i, aj = h . att_j
__global__ void k_node_attn(const float* __restrict__ h, const float* __restrict__ att,
                            float* __restrict__ ai, float* __restrict__ aj,
                            int N, int heads, int c) {
  int i = blockIdx.x * blockDim.x + threadIdx.x;
  if (i >= N * heads) return;
  int hh = i % heads;
  int n  = i / heads;
  int attStride = 2 * c + EDGEDIM;
  const float* hp  = h + ((long long)n * heads + hh) * c;
  const float* ati = att + hh * attStride;
  const float* atj = ati + c;
  float sa = 0.f, sb = 0.f;
  for (int k = 0; k < c; ++k) { float v = hp[k]; sa = fmaf(v, ati[k], sa); sb = fmaf(v, atj[k], sb); }
  ai[i] = sa; aj[i] = sb;
}

// per edge (incl. self-loops): gate MLP, attention logit, leaky-relu, segment max
__global__ void k_edge_alpha(const int* __restrict__ src, const int* __restrict__ dst,
                             const float* __restrict__ edge_attr, const float* __restrict__ loop_attr,
                             const float* __restrict__ ew1, const float* __restrict__ eb1,
                             const float* __restrict__ ew2, const float* __restrict__ eb2,
                             const float* __restrict__ att,
                             const float* __restrict__ ai, const float* __restrict__ aj,
                             float* __restrict__ alpha, unsigned* __restrict__ menc,
                             int E, int N, int heads, int c) {
  int e = blockIdx.x * blockDim.x + threadIdx.x;
  int E2 = E + N;
  if (e >= E2) return;
  int sn, dn;
  const float* eaPtr;
  if (e < E) { sn = src[e]; dn = dst[e]; eaPtr = edge_attr + (long long)e * EDGEDIM; }
  else       { sn = dn = e - E;          eaPtr = loop_attr + (long long)(e - E) * EDGEDIM; }
  float ea[EDGEDIM];
#pragma unroll
  for (int q = 0; q < EDGEDIM / 4; ++q) {
    v4f v = *(const v4f*)(eaPtr + 4 * q);
    ea[4 * q + 0] = v[0]; ea[4 * q + 1] = v[1]; ea[4 * q + 2] = v[2]; ea[4 * q + 3] = v[3];
  }
  // gate: sigmoid(relu(ea@ew1+eb1)@ew2+eb2); eaw = ea*gate
  float t[EDGEDIM / 2];
#pragma unroll
  for (int j = 0; j < EDGEDIM / 2; ++j) {
    float a = eb1[j];
#pragma unroll
    for (int k = 0; k < EDGEDIM; ++k) a = fmaf(ea[k], ew1[k * (EDGEDIM / 2) + j], a);
    t[j] = fmaxf(a, 0.f);
  }
#pragma unroll
  for (int o = 0; o < EDGEDIM; ++o) {
    float a = eb2[o];
#pragma unroll
    for (int j = 0; j < EDGEDIM / 2; ++j) a = fmaf(t[j], ew2[j * EDGEDIM + o], a);
    ea[o] *= 1.f / (1.f + expf(-a));
  }
  int attStride = 2 * c + EDGEDIM;
  for (int hh = 0; hh < heads; ++hh) {
    const float* ae = att + hh * attStride + 2 * c;
    float g = ai[dn * heads + hh] + aj[sn * heads + hh];
#pragma unroll
    for (int k = 0; k < EDGEDIM; ++k) g = fmaf(ea[k], ae[k], g);
    g = (g >= 0.f) ? g : 0.2f * g;                 // leaky_relu(0.2)
    alpha[(long long)e * heads + hh] = g;
    atomicMax(&menc[dn * heads + hh], enc_f32(g));
  }
}

// per (edge, head): e = exp(a - m[dst]); s[dst] += e; acc[dst] += e * h[src]
__global__ void k_edge_aggr(const int* __restrict__ src, const int* __restrict__ dst,
                            const float* __restrict__ h, const float* __restrict__ alpha,
                            const unsigned* __restrict__ menc,
                            float* __restrict__ ssum, float* __restrict__ acc,
                            int E, int N, int heads, int c) {
  int i = blockIdx.x * blockDim.x + threadIdx.x;
  int E2 = E + N;
  if (i >= E2 * heads) return;
  int hh = i % heads;
  int e  = i / heads;
  int sn, dn;
  if (e < E) { sn = src[e]; dn = dst[e]; } else { sn = dn = e - E; }
  float m  = dec_f32(menc[dn * heads + hh]);
  float ex = expf(alpha[i] - m);
  atomicAdd(&ssum[dn * heads + hh], ex);
  const float* hs = h   + ((long long)sn * heads + hh) * c;
  float*       ac = acc + ((long long)dn * heads + hh) * c;
  for (int k = 0; k < c; ++k) atomicAdd(&ac[k], ex * hs[k]);
}

// per (node, feature): normalize, +bias, optional BN + ELU
__global__ void k_finalize(const float* __restrict__ acc, const float* __restrict__ ssum,
                           const float* __restrict__ bias,
                           const float* __restrict__ gamma, const float* __restrict__ beta,
                           const float* __restrict__ mean,  const float* __restrict__ var,
                           float* __restrict__ xout, int N, int heads, int c, int doBnElu) {
  int i = blockIdx.x * blockDim.x + threadIdx.x;
  int dim = heads * c;
  if (i >= N * dim) return;
  int j  = i % dim;
  int n  = i / dim;
  int hh = j / c;
  float v = acc[i] / (ssum[n * heads + hh] + 1e-16f) + bias[j];
  if (doBnElu) {
    v = (v - mean[j]) * rsqrtf(var[j] + 1e-5f) * gamma[j] + beta[j];
    v = (v > 0.f) ? v : expm1f(v);                 // ELU
  }
  xout[i] = v;
}

static inline int cdiv(long long a, long long b) { return (int)((a + b - 1) / b); }

static void run_layer(hipStream_t stream, const float* xin, int inDim,
                      const float* W, const float* ew1, const float* eb1,
                      const float* ew2, const float* eb2,
                      const float* att, const float* bias,
                      int heads, int c,
                      const int* src, const int* dst,
                      const float* edge_attr, const float* loop_attr,
                      int N, int E,
                      float* hbuf, float* accb, float* ssum, unsigned* menc,
                      float* ai, float* aj, float* alpha,
                      const float* gamma, const float* beta,
                      const float* mean, const float* var,
                      float* xout) {
  const int TB = 256;
  const int M = heads * c;
  const int E2 = E + N;
  // dense transform (WMMA f32): one wave per 16x16 tile, 8 waves per block
  const int tiles = (N / 16) * (M / 16);
  k_gemm_wmma<<<cdiv(tiles, 8), 256, 0, stream>>>(xin, W, hbuf, N, inDim, M);
  // re-zero accumulators every call (workspace is poisoned by harness)
  k_fill_f32<<<cdiv((long long)N * M, TB), TB, 0, stream>>>(accb, 0.f, N * M);
  k_fill_f32<<<cdiv((long long)N * heads, TB), TB, 0, stream>>>(ssum, 0.f, N * heads);
  k_fill_u32<<<cdiv((long long)N * heads, TB), TB, 0, stream>>>(menc, 0u, N * heads);
  k_node_attn<<<cdiv((long long)N * heads, TB), TB, 0, stream>>>(hbuf, att, ai, aj, N, heads, c);
  k_edge_alpha<<<cdiv(E2, TB), TB, 0, stream>>>(src, dst, edge_attr, loop_attr,
                                                ew1, eb1, ew2, eb2, att, ai, aj,
                                                alpha, menc, E, N, heads, c);
  k_edge_aggr<<<cdiv((long long)E2 * heads, TB), TB, 0, stream>>>(src, dst, hbuf, alpha, menc,
                                                                  ssum, accb, E, N, heads, c);
  k_finalize<<<cdiv((long long)N * M, TB), TB, 0, stream>>>(accb, ssum, bias,
                                                            gamma, beta, mean, var,
                                                            xout, N, heads, c,
                                                            gamma != nullptr ? 1 : 0);
}

extern "C" void kernel_launch(void* const* d_in, const int* in_sizes, int n_in,
                              void* d_out, int out_size, void* d_ws, size_t ws_size,
                              hipStream_t stream) {
  const float* x         = (const float*)d_in[0];
  const int*   edge_idx  = (const int*)d_in[1];
  const float* edge_attr = (const float*)d_in[2];
  const int N = in_sizes[0] / 128;
  const int E = in_sizes[1] / 2;
  const int* src = edge_idx;
  const int* dst = edge_idx + E;

  const float *g1_W  = (const float*)d_in[3],  *g1_ew1 = (const float*)d_in[4];
  const float *g1_eb1= (const float*)d_in[5],  *g1_ew2 = (const float*)d_in[6];
  const float *g1_eb2= (const float*)d_in[7],  *g1_att = (const float*)d_in[8];
  const float *g1_b  = (const float*)d_in[9];
  const float *g2_W  = (const float*)d_in[10], *g2_ew1 = (const float*)d_in[11];
  const float *g2_eb1= (const float*)d_in[12], *g2_ew2 = (const float*)d_in[13];
  const float *g2_eb2= (const float*)d_in[14], *g2_att = (const float*)d_in[15];
  const float *g2_b  = (const float*)d_in[16];
  const float *g3_W  = (const float*)d_in[17], *g3_ew1 = (const float*)d_in[18];
  const float *g3_eb1= (const float*)d_in[19], *g3_ew2 = (const float*)d_in[20];
  const float *g3_eb2= (const float*)d_in[21], *g3_att = (const float*)d_in[22];
  const float *g3_b  = (const float*)d_in[23];
  const float *bn1_g = (const float*)d_in[24], *bn1_b = (const float*)d_in[25];
  const float *bn1_m = (const float*)d_in[26], *bn1_v = (const float*)d_in[27];
  const float *bn2_g = (const float*)d_in[28], *bn2_b = (const float*)d_in[29];
  const float *bn2_m = (const float*)d_in[30], *bn2_v = (const float*)d_in[31];

  // workspace layout (floats); total ~ N*817 + (E+N)*4  (~171 MB)
  float* ws        = (float*)d_ws;
  float* loop_attr = ws; ws += (size_t)N * EDGEDIM;
  float* cnt       = ws; ws += (size_t)N;            // must follow loop_attr (joint zero)
  float* hbuf      = ws; ws += (size_t)N * 256;
  float* accb      = ws; ws += (size_t)N * 256;
  float* xbuf      = ws; ws += (size_t)N * 256;
  float* ssum      = ws; ws += (size_t)N * 4;
  unsigned* menc   = (unsigned*)ws; ws += (size_t)N * 4;
  float* ai        = ws; ws += (size_t)N * 4;
  float* aj        = ws; ws += (size_t)N * 4;
  float* alpha     = ws; ws += (size_t)(E + N) * 4;
  (void)ws_size; (void)n_in; (void)out_size;

  const int TB = 256;
  // self-loop edge attributes (graph-only; recomputed every call for determinism)
  k_fill_f32<<<cdiv((long long)N * (EDGEDIM + 1), TB), TB, 0, stream>>>(loop_attr, 0.f, N * (EDGEDIM + 1));
  k_degree<<<cdiv(E, TB), TB, 0, stream>>>(dst, edge_attr, cnt, loop_attr, E);
  k_loop_attr<<<cdiv((long long)N * EDGEDIM, TB), TB, 0, stream>>>(loop_attr, cnt, N);

  // layer 1: 128 -> 4x64 (concat 256), BN1 + ELU
  run_layer(stream, x, 128, g1_W, g1_ew1, g1_eb1, g1_ew2, g1_eb2, g1_att, g1_b,
            4, 64, src, dst, edge_attr, loop_attr, N, E,
            hbuf, accb, ssum, menc, ai, aj, alpha,
            bn1_g, bn1_b, bn1_m, bn1_v, xbuf);
  // layer 2: 256 -> 4x32 (concat 128), BN2 + ELU   (xbuf in/out is stream-safe)
  run_layer(stream, xbuf, 256, g2_W, g2_ew1, g2_eb1, g2_ew2, g2_eb2, g2_att, g2_b,
            4, 32, src, dst, edge_attr, loop_attr, N, E,
            hbuf, accb, ssum, menc, ai, aj, alpha,
            bn2_g, bn2_b, bn2_m, bn2_v, xbuf);
  // layer 3: 128 -> 1x64 (mean over 1 head == identity), no BN/ELU -> d_out
  run_layer(stream, xbuf, 128, g3_W, g3_ew1, g3_eb1, g3_ew2, g3_eb2, g3_att, g3_b,
            1, 64, src, dst, edge_attr, loop_attr, N, E,
            hbuf, accb, ssum, menc, ai, aj, alpha,
            nullptr, nullptr, nullptr, nullptr, (float*)d_out);
}